// EmbLayer_10325101379594
// MI455X (gfx1250) — compile-verified
//
#include <hip/hip_runtime.h>
#include <hip/hip_bf16.h>
#include <stdint.h>

#define N_CELL 50000
#define N_GENE 50000
#define DIM    256
#define EPS    1e-5f

typedef float v2f __attribute__((ext_vector_type(2)));
typedef float v8f __attribute__((ext_vector_type(8)));

// ---------------------------------------------------------------------------
// Degree: deg_src[src[e]] += 1 ; deg_dst[dst[e]] += 1
// ---------------------------------------------------------------------------
__global__ void degree_kernel(const int* __restrict__ src,
                              const int* __restrict__ dst,
                              float* __restrict__ deg_src,
                              float* __restrict__ deg_dst, int E) {
    int i = blockIdx.x * blockDim.x + threadIdx.x;
    if (i >= E) return;
    atomicAdd(&deg_src[src[i]], 1.0f);
    atomicAdd(&deg_dst[dst[i]], 1.0f);
}

// ---------------------------------------------------------------------------
// Edge scatter: agg[dst[e]] += x_src[src[e]] * rsqrt(max(deg_out[src[e]],1))
// 64 threads per edge (float4 per thread) -> coalesced 512B gathers per wave.
// ---------------------------------------------------------------------------
__global__ void scatter_kernel(const float* __restrict__ x_src,
                               const int* __restrict__ src,
                               const int* __restrict__ dst,
                               const float* __restrict__ deg_out,
                               float* __restrict__ agg, int E) {
    long long t = (long long)blockIdx.x * blockDim.x + threadIdx.x;
    int e = (int)(t >> 6);
    int c = (int)(t & 63);
    if (e >= E) return;
    int s = src[e];
    int d = dst[e];
    float rs = rsqrtf(fmaxf(deg_out[s], 1.0f));
    const float4 v = ((const float4*)(x_src + (size_t)s * DIM))[c];
    float* o = agg + (size_t)d * DIM + c * 4;
    atomicAdd(o + 0, v.x * rs);
    atomicAdd(o + 1, v.y * rs);
    atomicAdd(o + 2, v.z * rs);
    atomicAdd(o + 3, v.w * rs);
}

// ---------------------------------------------------------------------------
// WMMA f32 16x16x4 GEMM pass:  C += diag(rsqrt(deg_in)) * A * W
// A fragment (16x4, ISA layout): lanes 0-15 M=0..15 half0, lanes 16-31 half1;
//   vreg0 -> K = k + 2*half, vreg1 -> K = k + 2*half + 1
// B fragment (4x16) mirrored: vreg0 -> row K = k + 2*half, vreg1 -> K+1,
//   N = (lane&15)
// ---------------------------------------------------------------------------
__device__ __forceinline__
v8f wmma_pass(const float* __restrict__ A, const float* __restrict__ deg_in,
              const float* __restrict__ W, int row0, int col0, int lane, v8f c) {
    const int m    = lane & 15;
    const int half = lane >> 4;
    const int row  = row0 + m;
    const float rs = rsqrtf(fmaxf(deg_in[row], 1.0f));
    const float* __restrict__ arow = A + (size_t)row * DIM;
    const int n = col0 + m;
    #pragma unroll 4
    for (int k = 0; k < DIM; k += 4) {
        const int ka = k + 2 * half;
        v2f a, b;
        a.x = arow[ka]     * rs;
        a.y = arow[ka + 1] * rs;
        b.x = W[(size_t)ka * DIM + n];
        b.y = W[(size_t)(ka + 1) * DIM + n];
        __builtin_prefetch(arow + k + 8, 0, 1);
        c = __builtin_amdgcn_wmma_f32_16x16x4_f32(
                /*neg_a=*/false, a, /*neg_b=*/false, b,
                /*c_mod=*/(short)0, c, /*reuse_a=*/false, /*reuse_b=*/false);
    }
    return c;
}

// One 16x16 output tile per wave; 4 waves per block.
// Optional second (A2,deg2,W2) accumulated into the same WMMA accumulator
// (fuses the two gene-side relations: saves one full H read/write pass).
__global__ void wmma_gemm_kernel(const float* __restrict__ A1,
                                 const float* __restrict__ deg1,
                                 const float* __restrict__ W1,
                                 const float* __restrict__ A2,
                                 const float* __restrict__ deg2,
                                 const float* __restrict__ W2,
                                 float* __restrict__ H, int nrows) {
    const int wave = threadIdx.x >> 5;
    const int lane = threadIdx.x & 31;
    const int row0 = blockIdx.x * 16;
    const int col0 = (blockIdx.y * 4 + wave) * 16;
    if (row0 >= nrows) return;            // uniform per block -> EXEC stays full

    v8f c = {};
    c = wmma_pass(A1, deg1, W1, row0, col0, lane, c);
    if (A2) c = wmma_pass(A2, deg2, W2, row0, col0, lane, c);

    const int m    = lane & 15;
    const int half = lane >> 4;
    #pragma unroll
    for (int i = 0; i < 8; ++i)           // D layout: vreg i -> row M = i + 8*half
        H[(size_t)(row0 + i + 8 * half) * DIM + col0 + m] = c[i];
}

// ---------------------------------------------------------------------------
// LayerNorm(affine) + bias + ReLU, one row (D=256) per 256-thread block.
// ---------------------------------------------------------------------------
__global__ void ln_relu_kernel(const float* __restrict__ H,
                               const float* __restrict__ gamma,
                               const float* __restrict__ beta,
                               const float* __restrict__ bias,
                               float* __restrict__ out, int nrows) {
    const int row = blockIdx.x;
    const int t   = threadIdx.x;
    const float v = H[(size_t)row * DIM + t];

    float s  = v;
    float sq = v * v;
    #pragma unroll
    for (int o = 16; o > 0; o >>= 1) {    // wave32 all-reduce
        s  += __shfl_xor(s,  o, 32);
        sq += __shfl_xor(sq, o, 32);
    }
    __shared__ float ls[8], lq[8];
    const int wid = t >> 5;
    if ((t & 31) == 0) { ls[wid] = s; lq[wid] = sq; }
    __syncthreads();
    __shared__ float smu, srv;
    if (t == 0) {
        float S = 0.f, Q = 0.f;
        #pragma unroll
        for (int i = 0; i < 8; ++i) { S += ls[i]; Q += lq[i]; }
        const float mu  = S * (1.0f / DIM);
        const float var = Q * (1.0f / DIM) - mu * mu;
        smu = mu;
        srv = rsqrtf(var + EPS);
    }
    __syncthreads();
    const float y = (v - smu) * srv * gamma[t] + beta[t] + bias[t];
    out[(size_t)row * DIM + t] = fmaxf(y, 0.0f);
}

// ---------------------------------------------------------------------------
extern "C" void kernel_launch(void* const* d_in, const int* in_sizes, int n_in,
                              void* d_out, int out_size, void* d_ws, size_t ws_size,
                              hipStream_t stream) {
    const float* x_cell     = (const float*)d_in[0];
    const float* x_gene     = (const float*)d_in[1];
    const float* W_cg       = (const float*)d_in[2];
    const float* W_gc       = (const float*)d_in[3];
    const float* W_gg       = (const float*)d_in[4];
    const float* gamma_cell = (const float*)d_in[5];
    const float* beta_cell  = (const float*)d_in[6];
    const float* gamma_gene = (const float*)d_in[7];
    const float* beta_gene  = (const float*)d_in[8];
    const float* b_cell     = (const float*)d_in[9];
    const float* b_gene     = (const float*)d_in[10];
    const int*   src_cg     = (const int*)d_in[11];
    const int*   dst_cg     = (const int*)d_in[12];
    const int*   src_gc     = (const int*)d_in[13];
    const int*   dst_gc     = (const int*)d_in[14];
    const int*   src_gg     = (const int*)d_in[15];
    const int*   dst_gg     = (const int*)d_in[16];
    const int E = in_sizes[11];

    // Workspace carve-up (~257 MB)
    float* ws = (float*)d_ws;
    float* deg_out_cg = ws;                                   // N_CELL
    float* deg_in_cg  = deg_out_cg + N_CELL;                  // N_GENE
    float* deg_out_gg = deg_in_cg  + N_GENE;                  // N_GENE
    float* deg_in_gg  = deg_out_gg + N_GENE;                  // N_GENE
    float* deg_out_gc = deg_in_gg  + N_GENE;                  // N_GENE
    float* deg_in_gc  = deg_out_gc + N_GENE;                  // N_CELL
    float* agg_cg = deg_in_gc + N_CELL;                       // N_GENE*DIM
    float* agg_gg = agg_cg + (size_t)N_GENE * DIM;            // N_GENE*DIM
    float* agg_gc = agg_gg + (size_t)N_GENE * DIM;            // N_CELL*DIM
    float* h_gene = agg_gc + (size_t)N_CELL * DIM;            // N_GENE*DIM
    float* h_cell = h_gene + (size_t)N_GENE * DIM;            // N_CELL*DIM

    // Zero degrees + agg accumulators (contiguous from ws start)
    const size_t zero_bytes =
        ((size_t)(2 * N_CELL + 4 * N_GENE) + 3ull * N_GENE * DIM) * sizeof(float);
    hipMemsetAsync(d_ws, 0, zero_bytes, stream);

    // Degrees
    {
        dim3 b(256), g((E + 255) / 256);
        degree_kernel<<<g, b, 0, stream>>>(src_cg, dst_cg, deg_out_cg, deg_in_cg, E);
        degree_kernel<<<g, b, 0, stream>>>(src_gg, dst_gg, deg_out_gg, deg_in_gg, E);
        degree_kernel<<<g, b, 0, stream>>>(src_gc, dst_gc, deg_out_gc, deg_in_gc, E);
    }

    // Normalized gather + atomic scatter (64 threads/edge)
    {
        long long total = (long long)E * 64;
        dim3 b(256), g((unsigned)((total + 255) / 256));
        scatter_kernel<<<g, b, 0, stream>>>(x_cell, src_cg, dst_cg, deg_out_cg, agg_cg, E);
        scatter_kernel<<<g, b, 0, stream>>>(x_gene, src_gg, dst_gg, deg_out_gg, agg_gg, E);
        scatter_kernel<<<g, b, 0, stream>>>(x_gene, src_gc, dst_gc, deg_out_gc, agg_gc, E);
    }

    // WMMA f32 GEMMs (4 waves/block, 16x16 tile per wave)
    {
        dim3 b(128);
        dim3 g_gene(N_GENE / 16, DIM / 16 / 4);
        dim3 g_cell(N_CELL / 16, DIM / 16 / 4);
        // h_gene = diag(rs)agg_cg @ W_cg + diag(rs)agg_gg @ W_gg  (fused)
        wmma_gemm_kernel<<<g_gene, b, 0, stream>>>(agg_cg, deg_in_cg, W_cg,
                                                   agg_gg, deg_in_gg, W_gg,
                                                   h_gene, N_GENE);
        wmma_gemm_kernel<<<g_cell, b, 0, stream>>>(agg_gc, deg_in_gc, W_gc,
                                                   nullptr, nullptr, nullptr,
                                                   h_cell, N_CELL);
    }

    // LayerNorm + bias + ReLU; output stacked [out_cell, out_gene]
    {
        float* out_cell = (float*)d_out;
        float* out_gene = out_cell + (size_t)N_CELL * DIM;
        ln_relu_kernel<<<dim3(N_CELL), dim3(DIM), 0, stream>>>(
            h_cell, gamma_cell, beta_cell, b_cell, out_cell, N_CELL);
        ln_relu_kernel<<<dim3(N_GENE), dim3(DIM), 0, stream>>>(
            h_gene, gamma_gene, beta_gene, b_gene, out_gene, N_GENE);
    }
}